// GCN_16509854285962
// MI455X (gfx1250) — compile-verified
//
#include <hip/hip_runtime.h>
#include <hip/hip_bf16.h>

typedef float v2f __attribute__((ext_vector_type(2)));
typedef float v8f __attribute__((ext_vector_type(8)));

#define BN_EPS 1e-5f

__device__ __forceinline__ void atomicAddF32(float* p, float v) {
  unsafeAtomicAdd(p, v);  // hardware GLOBAL_ATOMIC_ADD_F32
}

// ---------------- small utility kernels ----------------

__global__ void k_fill(float* __restrict__ p, float v, int n) {
  int i = blockIdx.x * blockDim.x + threadIdx.x;
  if (i < n) p[i] = v;
}

__global__ void k_deg_count(const long long* __restrict__ dst, float* __restrict__ deg, int E) {
  int i = blockIdx.x * blockDim.x + threadIdx.x;
  if (i < E) atomicAddF32(&deg[(int)dst[i]], 1.0f);
}

__global__ void k_rsqrt_inplace(float* __restrict__ p, int n) {
  int i = blockIdx.x * blockDim.x + threadIdx.x;
  if (i < n) p[i] = rsqrtf(p[i]);
}

// out[i,c] = h[i,c] * dinv[i]^2 + bias[c]   (self-loop contribution + bias)
__global__ void k_init_out(const float* __restrict__ h, const float* __restrict__ dinv,
                           const float* __restrict__ bias, float* __restrict__ out,
                           int N, int C) {
  int t = blockIdx.x * blockDim.x + threadIdx.x;
  if (t >= N * C) return;
  int i = t / C, c = t - i * C;
  float dv = dinv[i];
  out[t] = fmaf(h[t], dv * dv, bias[c]);
}

// ---------------- edge scatter-add (one wave per edge) ----------------

template <int C>
__global__ __launch_bounds__(256) void k_edge_agg(const float* __restrict__ h,
                                                  const long long* __restrict__ src,
                                                  const long long* __restrict__ dst,
                                                  const float* __restrict__ dinv,
                                                  float* __restrict__ out, int E) {
  int e = blockIdx.x * (blockDim.x >> 5) + (threadIdx.x >> 5);  // wave-uniform
  if (e >= E) return;
  int lane = threadIdx.x & 31;
  int s = (int)src[e];
  int d = (int)dst[e];
  float nrm = dinv[s] * dinv[d];
  constexpr int PER = C / 32;
  const float* hp = h + (size_t)s * C + lane * PER;
  float* op = out + (size_t)d * C + lane * PER;
  float v[PER];
#pragma unroll
  for (int j = 0; j < PER; ++j) v[j] = hp[j];
#pragma unroll
  for (int j = 0; j < PER; ++j) atomicAddF32(op + j, v[j] * nrm);
}

// ---------------- WMMA f32 GEMM: Y[N,COUT] = X[N,128] @ W[128,COUT] ----------------
//
// LDS layout of W: k-row pairs interleaved by column, element (r,c) at
//   sW[(r/2)*PST + 2*c + (r&1)]
// so a B fragment {W[r][col], W[r+1][col]} is one contiguous ds_load_b64.
// PST = 2*COUT + 32 dwords => PST % 64 == 32, so the two lane halves
// (which read pair-rows r/2 and r/2+1) land on disjoint bank halves:
// lanes 0-15 -> banks 0-31, lanes 16-31 -> banks 32-63. Conflict-free b64.

template <int COUT>
__global__ __launch_bounds__(256) void gemm_wmma(const float* __restrict__ X,
                                                 const float* __restrict__ Wm,
                                                 float* __restrict__ Y, int N) {
  constexpr int K = 128;
  constexpr int NT = COUT / 16;
  constexpr int PST = COUT * 2 + 32;  // dwords per k-pair row
  __shared__ float sW[(K / 2) * PST];

  // cooperative stage of W into paired LDS layout
  for (int p = threadIdx.x; p < (K / 2) * COUT; p += 256) {
    int rp = p / COUT, c = p - rp * COUT;
    float2 w2;
    w2.x = Wm[(2 * rp) * COUT + c];
    w2.y = Wm[(2 * rp + 1) * COUT + c];
    *(float2*)&sW[rp * PST + 2 * c] = w2;
  }
  __syncthreads();

  int wave = threadIdx.x >> 5;
  int lane = threadIdx.x & 31;
  int rowBlock = (blockIdx.x * 8 + wave) * 16;
  if (rowBlock >= N) return;  // wave-uniform

  int m = lane & 15;
  int kh = (lane >> 4) * 2;   // lanes 16-31 carry K+2, K+3
  int rowA = rowBlock + m;
  if (rowA >= N) rowA = N - 1;  // clamp (stores are guarded)
  const float* xr = X + (size_t)rowA * K;

  v8f acc[NT];
#pragma unroll
  for (int t = 0; t < NT; ++t) acc[t] = v8f{0.f, 0.f, 0.f, 0.f, 0.f, 0.f, 0.f, 0.f};

#pragma unroll 4
  for (int k = 0; k < K; k += 4) {
    v2f a;
    a.x = xr[k + kh];
    a.y = xr[k + kh + 1];
    const float2* brow = (const float2*)&sW[((k + kh) >> 1) * PST];
#pragma unroll
    for (int t = 0; t < NT; ++t) {
      float2 bb = brow[t * 16 + m];  // single ds_load_b64: {W[r][col], W[r+1][col]}
      v2f b;
      b.x = bb.x;
      b.y = bb.y;
      acc[t] = __builtin_amdgcn_wmma_f32_16x16x4_f32(false, a, false, b, (short)0,
                                                     acc[t], false, false);
    }
  }

  // store: VGPR r -> row rowBlock + r (+8 for upper lane half), col = t*16 + m
  int rbase = rowBlock + (lane >> 4) * 8;
#pragma unroll
  for (int t = 0; t < NT; ++t) {
    int col = t * 16 + m;
#pragma unroll
    for (int r = 0; r < 8; ++r) {
      int row = rbase + r;
      if (row < N) Y[(size_t)row * COUT + col] = acc[t][r];
    }
  }
}

// ---------------- BatchNorm ----------------

__global__ void k_bn_stats(const float* __restrict__ x, float* __restrict__ sums,
                           float* __restrict__ sqs, int N, int C) {
  int c = threadIdx.x;  // blockDim.x == C
  float s = 0.f, q = 0.f;
  for (int i = blockIdx.x; i < N; i += gridDim.x) {
    float v = x[(size_t)i * C + c];
    s += v;
    q += v * v;
  }
  atomicAddF32(&sums[c], s);
  atomicAddF32(&sqs[c], q);
}

__global__ void k_bn_finalize(const float* __restrict__ sums, const float* __restrict__ sqs,
                              const float* __restrict__ gamma, const float* __restrict__ beta,
                              float* __restrict__ scale, float* __restrict__ shift,
                              int N, int C) {
  int c = threadIdx.x;
  if (c >= C) return;
  float inv_n = 1.0f / (float)N;
  float mu = sums[c] * inv_n;
  float var = sqs[c] * inv_n - mu * mu;
  float sc = gamma[c] * rsqrtf(var + BN_EPS);
  scale[c] = sc;
  shift[c] = beta[c] - mu * sc;
}

__global__ void k_bn_apply(float* __restrict__ x, const float* __restrict__ scale,
                           const float* __restrict__ shift, int total, int Cmask, int relu) {
  int t = blockIdx.x * blockDim.x + threadIdx.x;
  if (t >= total) return;
  int c = t & Cmask;  // C is a power of two
  float y = fmaf(x[t], scale[c], shift[c]);
  if (relu) y = fmaxf(y, 0.f);
  x[t] = y;
}

// ---------------- host orchestration ----------------

static void run_bn(float* buf, const float* gamma, const float* beta, float* sums,
                   float* sqs, float* scale, float* shift, int N, int C, int relu,
                   hipStream_t stream) {
  k_fill<<<1, 256, 0, stream>>>(sums, 0.f, 256);  // sums+sqs contiguous
  k_bn_stats<<<1024, C, 0, stream>>>(buf, sums, sqs, N, C);
  k_bn_finalize<<<1, C, 0, stream>>>(sums, sqs, gamma, beta, scale, shift, N, C);
  int total = N * C;
  k_bn_apply<<<(total + 255) / 256, 256, 0, stream>>>(buf, scale, shift, total, C - 1, relu);
}

extern "C" void kernel_launch(void* const* d_in, const int* in_sizes, int n_in,
                              void* d_out, int out_size, void* d_ws, size_t ws_size,
                              hipStream_t stream) {
  const float* x = (const float*)d_in[0];
  const long long* ei = (const long long*)d_in[1];
  const float* W1 = (const float*)d_in[2];
  const float* b1 = (const float*)d_in[3];
  const float* g1 = (const float*)d_in[4];
  const float* be1 = (const float*)d_in[5];
  const float* W2 = (const float*)d_in[6];
  const float* b2 = (const float*)d_in[7];
  const float* g2 = (const float*)d_in[8];
  const float* be2 = (const float*)d_in[9];
  const float* W3 = (const float*)d_in[10];
  const float* b3 = (const float*)d_in[11];
  const float* g3 = (const float*)d_in[12];
  const float* be3 = (const float*)d_in[13];

  const int Cin = 128, H = 128, Cout = 64;
  int N = in_sizes[0] / Cin;       // 100000
  int E = in_sizes[1] / 2;         // 1600000
  const long long* src = ei;
  const long long* dstv = ei + E;
  float* outp = (float*)d_out;

  // workspace layout
  float* ws = (float*)d_ws;
  float* dinv = ws;                         // N
  float* sums = dinv + N;                   // 128
  float* sqs = sums + 128;                  // 128
  float* scale = sqs + 128;                 // 128
  float* shift = scale + 128;               // 128
  float* gbuf = shift + 128;                // N*128 (GEMM output h)
  float* abuf = gbuf + (size_t)N * 128;     // N*128 (aggregated activations)

  int gN = (N + 255) / 256;
  int gE = (E + 255) / 256;
  int gEdgeWave = (E + 7) / 8;                        // 8 waves per 256-thread block
  int gGemm = (((N + 15) / 16) + 7) / 8;              // 8 row-tiles per block

  // --- degree -> dinv (self-loop included: deg starts at 1) ---
  k_fill<<<gN, 256, 0, stream>>>(dinv, 1.0f, N);
  k_deg_count<<<gE, 256, 0, stream>>>(dstv, dinv, E);
  k_rsqrt_inplace<<<gN, 256, 0, stream>>>(dinv, N);

  // --- layer 1: GCNConv(x, W1, b1) -> BN -> ReLU ---
  gemm_wmma<128><<<gGemm, 256, 0, stream>>>(x, W1, gbuf, N);
  k_init_out<<<(N * H + 255) / 256, 256, 0, stream>>>(gbuf, dinv, b1, abuf, N, H);
  k_edge_agg<128><<<gEdgeWave, 256, 0, stream>>>(gbuf, src, dstv, dinv, abuf, E);
  run_bn(abuf, g1, be1, sums, sqs, scale, shift, N, H, 1, stream);

  // --- layer 2 ---
  gemm_wmma<128><<<gGemm, 256, 0, stream>>>(abuf, W2, gbuf, N);
  k_init_out<<<(N * H + 255) / 256, 256, 0, stream>>>(gbuf, dinv, b2, abuf, N, H);
  k_edge_agg<128><<<gEdgeWave, 256, 0, stream>>>(gbuf, src, dstv, dinv, abuf, E);
  run_bn(abuf, g2, be2, sums, sqs, scale, shift, N, H, 1, stream);

  // --- layer 3 (Cout=64), BN only, into d_out ---
  gemm_wmma<64><<<gGemm, 256, 0, stream>>>(abuf, W3, gbuf, N);
  k_init_out<<<(N * Cout + 255) / 256, 256, 0, stream>>>(gbuf, dinv, b3, outp, N, Cout);
  k_edge_agg<64><<<gEdgeWave, 256, 0, stream>>>(gbuf, src, dstv, dinv, outp, E);
  run_bn(outp, g3, be3, sums, sqs, scale, shift, N, Cout, 0, stream);
}